// ODOCSegEdgeGruGcn_50654844289082
// MI455X (gfx1250) — compile-verified
//
#include <hip/hip_runtime.h>

typedef __attribute__((ext_vector_type(2))) float v2f;
typedef __attribute__((ext_vector_type(8))) float v8f;

#define NN 8            // nodes
#define NB 4            // batch
#define CC 32           // channels
#define HW (96 * 96)    // spatial per (node,batch,channel) = 9216
#define TILES_PER_WAVE 4
#define WAVES_PER_BLOCK 4
// pixel tiles total = NB*HW/16 = 2304 ; grid.x = 2304/(4*4) = 144

static __device__ __forceinline__ v8f wmma4(v2f a, v2f b, v8f c) {
  // D(16x16,f32) = A(16x4,f32) x B(4x16,f32) + C
  return __builtin_amdgcn_wmma_f32_16x16x4_f32(false, a, false, b, (short)0, c,
                                               false, false);
}

static __device__ __forceinline__ v8f vzero8() {
  v8f v;
#pragma unroll
  for (int r = 0; r < 8; ++r) v[r] = 0.0f;
  return v;
}

// B-operand tile for one node: channels (K) x 16 pixels (N).
// Chunk kk covers K = 4kk..4kk+3. Lane (h,n): v.x -> (c=4kk+2h, pix n),
// v.y -> (c=4kk+2h+1, pix n). Channel stride in memory is HW.
static __device__ __forceinline__ void load_Btile(const float* __restrict__ src,
                                                  int node, int bb, int p,
                                                  int n, int h, v2f bt[8]) {
  const float* base = src + ((long)(node * NB + bb) * CC) * HW + p + n;
#pragma unroll
  for (int kk = 0; kk < 8; ++kk) {
    const int c0 = 4 * kk + 2 * h;
    v2f v;
    v.x = base[(long)c0 * HW];
    v.y = base[(long)(c0 + 1) * HW];
    bt[kk] = v;
  }
}

// C/D-layout tile for one node: VGPR r, lane (h,n) -> (c = 16m + 8h + r, pix n)
static __device__ __forceinline__ void load_Dtile(const float* __restrict__ src,
                                                  int node, int bb, int p,
                                                  int n, int h, v8f dt[2]) {
  const float* base = src + ((long)(node * NB + bb) * CC) * HW + p + n;
#pragma unroll
  for (int m = 0; m < 2; ++m)
#pragma unroll
    for (int r = 0; r < 8; ++r)
      dt[m][r] = base[(long)(16 * m + 8 * h + r) * HW];
}

// 32x32 weight (A-layout, resident in VGPRs) times B tile -> two 16x16 tiles
static __device__ __forceinline__ void gemm32(const v2f wa[2][8],
                                              const v2f bt[8], v8f out[2]) {
#pragma unroll
  for (int m = 0; m < 2; ++m)
#pragma unroll
    for (int kk = 0; kk < 8; ++kk) out[m] = wmma4(wa[m][kk], bt[kk], out[m]);
}

__global__ __launch_bounds__(WAVES_PER_BLOCK * 32) void gcn_step_kernel(
    const float* __restrict__ src, const float* __restrict__ Wedge,
    const float* __restrict__ bedge, float* __restrict__ dst) {
  const int lane = threadIdx.x & 31;
  const int wave = threadIdx.x >> 5;
  const int n = lane & 15;  // pixel within tile (N)
  const int h = lane >> 4;  // lane half (K / M split)
  const int i = blockIdx.y; // receiver node

  // ---- receiver weights, resident in registers (A-matrix layout) ----
  // Wd = W_edge[i][:, :32], W2 = Ws - Wd = W_edge[i][:, 32:] - Wd
  v2f wd[2][8], w2[2][8];
  const float* Wi = Wedge + (long)i * CC * (2 * CC);
#pragma unroll
  for (int m = 0; m < 2; ++m) {
    const float* row = Wi + (long)(16 * m + n) * (2 * CC);
#pragma unroll
    for (int kk = 0; kk < 8; ++kk) {
      const int c0 = 4 * kk + 2 * h;
      const float d0 = row[c0], d1 = row[c0 + 1];
      const float s0 = row[CC + c0], s1 = row[CC + c0 + 1];
      v2f a;
      a.x = d0; a.y = d1;
      wd[m][kk] = a;
      v2f b2;
      b2.x = s0 - d0; b2.y = s1 - d1;
      w2[m][kk] = b2;
    }
  }
  // bias in C/D layout
  v8f bias[2];
#pragma unroll
  for (int m = 0; m < 2; ++m)
#pragma unroll
    for (int r = 0; r < 8; ++r)
      bias[m][r] = bedge[i * CC + 16 * m + 8 * h + r];

  const int tile0 =
      (blockIdx.x * WAVES_PER_BLOCK + wave) * TILES_PER_WAVE;

  for (int t = 0; t < TILES_PER_WAVE; ++t) {
    const int q = (tile0 + t) * 16;  // flattened pixel index over NB*HW
    const int bb = q / HW;
    const int p = q - bb * HW;       // tiles never straddle HW (9216 % 16 == 0)

    // T = (Ws - Wd) @ nodes_i + b   (per receiver, reused for all senders)
    v2f bt[8];
    load_Btile(src, i, bb, p, n, h, bt);
    v8f T[2] = {bias[0], bias[1]};
    gemm32(w2, bt, T);

    v8f acc[2] = {vzero8(), vzero8()};

    for (int j = 0; j < NN; ++j) {
      if (j == i) continue;  // uniform branch: EXEC stays all-ones
      load_Btile(src, j, bb, p, n, h, bt);
      v8f P[2] = {T[0], T[1]};
      gemm32(wd, bt, P);     // P = Wd @ n_j + T  == edge pre-activation
      v8f dv[2];
      load_Dtile(src, j, bb, p, n, h, dv);  // n_j in C/D layout for the gate
#pragma unroll
      for (int m = 0; m < 2; ++m)
#pragma unroll
        for (int r = 0; r < 8; ++r)
          acc[m][r] += fmaxf(P[m][r], 0.0f) * dv[m][r];
    }

    // out_i = msg + n_i
    v8f self[2];
    load_Dtile(src, i, bb, p, n, h, self);
    float* obase = dst + ((long)(i * NB + bb) * CC) * HW + p + n;
#pragma unroll
    for (int m = 0; m < 2; ++m)
#pragma unroll
      for (int r = 0; r < 8; ++r)
        obase[(long)(16 * m + 8 * h + r) * HW] = acc[m][r] + self[m][r];
  }
}

extern "C" void kernel_launch(void* const* d_in, const int* in_sizes, int n_in,
                              void* d_out, int out_size, void* d_ws,
                              size_t ws_size, hipStream_t stream) {
  const float* nodes = (const float*)d_in[0];  // [8,4,32,96,96]
  const float* Wedge = (const float*)d_in[1];  // [8,32,64]
  const float* bedge = (const float*)d_in[2];  // [8,32]
  float* out = (float*)d_out;
  float* ws = (float*)d_ws;  // needs 8*4*32*96*96*4 = ~37.7 MB

  dim3 block(WAVES_PER_BLOCK * 32);
  dim3 grid((NB * HW / 16) / (WAVES_PER_BLOCK * TILES_PER_WAVE), NN);

  // 3 message-passing steps, ping-pong: in -> out -> ws -> out
  gcn_step_kernel<<<grid, block, 0, stream>>>(nodes, Wedge, bedge, out);
  gcn_step_kernel<<<grid, block, 0, stream>>>(out, Wedge, bedge, ws);
  gcn_step_kernel<<<grid, block, 0, stream>>>(ws, Wedge, bedge, out);
}